// DiffeomorphicLayer_79276506349717
// MI455X (gfx1250) — compile-verified
//
#include <hip/hip_runtime.h>

#define VOLD 128
#define VOLH 128
#define VOLW 128
#define TS   8            // output tile edge
#define HLO  4            // halo below
#define HHI  5            // halo above (floor(p)+1 with |disp|<=4)
#define EMAX (TS + HLO + HHI)          // 17
#define CH_STRIDE (VOLD * VOLH * VOLW) // 2097152 elements
#define B_STRIDE  (3 * CH_STRIDE)

typedef __attribute__((ext_vector_type(4))) unsigned int v4u;
typedef __attribute__((ext_vector_type(8))) int          v8i;
typedef __attribute__((ext_vector_type(4))) int          v4i;
typedef __attribute__((address_space(3))) float          ldsf;  // 32-bit LDS pointer

__global__ void init_flow(const float* __restrict__ vel, float* __restrict__ flow, int n4) {
    int i = blockIdx.x * blockDim.x + threadIdx.x;
    if (i < n4) {
        const float4* v = (const float4*)vel;
        float4* f = (float4*)flow;
        float4 a = v[i];
        a.x *= 0.0078125f; a.y *= 0.0078125f; a.z *= 0.0078125f; a.w *= 0.0078125f;
        f[i] = a;
    }
}

__launch_bounds__(512)
__global__ void step_kernel(const float* __restrict__ src, float* __restrict__ dst) {
    __shared__ float smem[3 * EMAX * EMAX * EMAX];   // 58,956 bytes

    const int tx0 = blockIdx.x * TS;
    const int ty0 = blockIdx.y * TS;
    const int bz  = blockIdx.z;
    const int b   = bz >> 4;            // 16 z-tiles per batch
    const int tz0 = (bz & 15) * TS;

    // source halo box clamped to the volume (so TDM reads stay in-bounds)
    const int sx = max(tx0 - HLO, 0);
    const int sy = max(ty0 - HLO, 0);
    const int sz = max(tz0 - HLO, 0);
    const int exh = min(tx0 + TS - 1 + HHI, VOLW - 1);
    const int eyh = min(ty0 + TS - 1 + HHI, VOLH - 1);
    const int ezh = min(tz0 + TS - 1 + HHI, VOLD - 1);
    const int lenx = exh - sx + 1;
    const int leny = eyh - sy + 1;
    const int lenz = ezh - sz + 1;
    const int csz  = lenx * leny * lenz;     // per-channel packed LDS elements

    if (threadIdx.x < 32) {
        // ---- Tensor DMA descriptor (D#), 4D tile: (lenx, leny, lenz, 3 channels) ----
        unsigned long long gaddr =
            (unsigned long long)(const void*)(src + (size_t)b * B_STRIDE +
                                              ((size_t)sz * VOLH + sy) * VOLW + sx);
        unsigned int ldsa = (unsigned int)(unsigned long long)(const void*)smem; // low 32b = LDS offset

        v4u g0; v8i g1; v4i g2; v4i g3; v8i gx;
        // group0: count=1, lds_addr, global_addr, type=2
        g0[0] = 1u;
        g0[1] = ldsa;
        g0[2] = (unsigned int)(gaddr & 0xFFFFFFFFull);
        g0[3] = (unsigned int)((gaddr >> 32) & 0x01FFFFFFull) | 0x80000000u;
        // group1: wg_mask=0, data_size=4B (code 2); big tensor dims (box pre-clamped);
        //         tile dims; strides 128 and 128*128
        g1[0] = (int)(2u << 16);                       // data_size=2 -> 4 bytes
        g1[1] = 0;                                     // ab_addr=0 | tensor_dim0[15:0]=0
        g1[2] = (int)0x00004000;                       // tensor_dim0[31:16]=0x4000 (1<<30) | dim1 lo=0
        g1[3] = (int)(0x00004000u | ((unsigned)lenx << 16)); // dim1 hi | tile_dim0
        g1[4] = (int)(((unsigned)leny) | ((unsigned)lenz << 16)); // tile_dim1 | tile_dim2
        g1[5] = (int)VOLW;                             // tensor_dim0_stride lo32 = 128
        g1[6] = (int)(((unsigned)(VOLH * VOLW) & 0xFFFFu) << 16); // stride0 hi=0 | stride1[15:0]
        g1[7] = (int)((unsigned)(VOLH * VOLW) >> 16);  // stride1[47:16]
        // group2: tensor_dim2 big, tensor_dim3=16, stride2 = channel stride, tile_dim3=3
        g2[0] = (int)0x40000000;
        g2[1] = 16;
        g2[2] = (int)CH_STRIDE;                        // 0x200000
        g2[3] = (int)(3u << 16);                       // stride2 hi=0 | tile_dim3=3
        // group3: dim4 unused
        g3[0] = 0; g3[1] = 0; g3[2] = 0; g3[3] = 0;
        // extra group (6-arg builtin form): zero-filled (single-descriptor op, count=1)
        gx[0] = 0; gx[1] = 0; gx[2] = 0; gx[3] = 0;
        gx[4] = 0; gx[5] = 0; gx[6] = 0; gx[7] = 0;

        __builtin_amdgcn_tensor_load_to_lds(g0, g1, g2, g3, gx, 0);
        __builtin_amdgcn_s_wait_tensorcnt(0);
    }
    __syncthreads();

    // Opaque 32-bit LDS base offset: keeps loads in addrspace(3) (ds_load) while
    // preventing the compiler from folding loads of "never-stored" shared memory.
    unsigned int base = (unsigned int)(unsigned long long)(const void*)smem;
    asm volatile("" : "+s"(base) :: "memory");
    ldsf* c0 = (ldsf*)base;
    ldsf* c1 = c0 + csz;
    ldsf* c2 = c0 + 2 * csz;

    const int ltid = threadIdx.x;            // 512 threads -> 8x8x8 voxels
    const int lx = ltid & 7;
    const int ly = (ltid >> 3) & 7;
    const int lz = ltid >> 6;
    const int x = tx0 + lx, y = ty0 + ly, z = tz0 + lz;

    const int own = ((z - sz) * leny + (y - sy)) * lenx + (x - sx);
    const float fz = c0[own], fy = c1[own], fx = c2[own];

    const float px = (float)x + fx;
    const float py = (float)y + fy;
    const float pz = (float)z + fz;

    const float x0f = floorf(px), y0f = floorf(py), z0f = floorf(pz);
    const int ix0 = (int)x0f, iy0 = (int)y0f, iz0 = (int)z0f;
    const float wx1 = px - x0f, wx0 = 1.0f - wx1;
    const float wy1 = py - y0f, wy0 = 1.0f - wy1;
    const float wz1 = pz - z0f, wz0 = 1.0f - wz1;

    float a0 = 0.0f, a1 = 0.0f, a2 = 0.0f;
#pragma unroll
    for (int k = 0; k < 8; ++k) {
        const int cx = ix0 + (k & 1);
        const int cy = iy0 + ((k >> 1) & 1);
        const int cz = iz0 + (k >> 2);
        const bool valid = (cx >= 0) && (cx < VOLW) && (cy >= 0) && (cy < VOLH) &&
                           (cz >= 0) && (cz < VOLD);
        float w = ((k & 1) ? wx1 : wx0) *
                  (((k >> 1) & 1) ? wy1 : wy0) *
                  ((k >> 2) ? wz1 : wz0);
        w = valid ? w : 0.0f;
        // box-relative coords; clamp for memory safety (masked lanes read garbage*0)
        const int bx = min(max(cx - sx, 0), lenx - 1);
        const int by = min(max(cy - sy, 0), leny - 1);
        const int bw = min(max(cz - sz, 0), lenz - 1);
        const int idx = (bw * leny + by) * lenx + bx;
        a0 = fmaf(w, c0[idx], a0);
        a1 = fmaf(w, c1[idx], a1);
        a2 = fmaf(w, c2[idx], a2);
    }

    const size_t ob = (size_t)b * B_STRIDE + ((size_t)z * VOLH + y) * VOLW + x;
    dst[ob]                 = fz + a0;   // channel 0 (z-flow)
    dst[ob + CH_STRIDE]     = fy + a1;   // channel 1 (y-flow)
    dst[ob + 2 * CH_STRIDE] = fx + a2;   // channel 2 (x-flow)
}

extern "C" void kernel_launch(void* const* d_in, const int* in_sizes, int n_in,
                              void* d_out, int out_size, void* d_ws, size_t ws_size,
                              hipStream_t stream) {
    const float* vel = (const float*)d_in[0];
    float* out  = (float*)d_out;
    float* bufA = (float*)d_ws;          // 50.3 MB scratch flow buffer

    const int n  = 2 * 3 * VOLD * VOLH * VOLW;
    const int n4 = n / 4;
    init_flow<<<(n4 + 255) / 256, 256, 0, stream>>>(vel, bufA, n4);

    dim3 grid(VOLW / TS, VOLH / TS, (VOLD / TS) * 2 /*batch*/);
    float* bufs[2] = { bufA, out };
    for (int i = 0; i < 7; ++i) {
        const float* s = bufs[i & 1];        // even: d_ws, odd: d_out
        float*       d = bufs[(i + 1) & 1];
        step_kernel<<<grid, 512, 0, stream>>>(s, d);
    }
    // 7 steps (odd) => final result lands in d_out
}